// FusionNet_10995116278260
// MI455X (gfx1250) — compile-verified
//
#include <hip/hip_runtime.h>
#include <math.h>

typedef __attribute__((ext_vector_type(2))) float v2f;
typedef __attribute__((ext_vector_type(8))) float v8f;

#define C64   64
#define W192  192
#define H192  192
#define PLANE 36864   // H*W
#define LDQ   196     // stride for [64][192] buffers (mod 64 == 4 -> conflict-free frags)
#define LDA   196     // stride for attn [192][192]
#define LDW   65      // stride for staged 64x64 weights (odd -> conflict-free)

// LDS float offsets (single dynamic-shared arena, phased reuse)
#define OFF_ATT   0        // attn [192][196] = 37632 floats (phase >= 3)
#define OFF_LNX   0        // lnx  [64][196]  = 12544 floats (phase 1-2, inside attn region)
#define OFF_LNY   12544    // lny  (phase 1-2)
#define OFF_QL    37632    // Ql, later Vl
#define OFF_QR    50176    // Qr, later Vr
#define OFF_RMAX  62720
#define OFF_RSUM  62912
#define OFF_CMAX  63104
#define OFF_CSUM  63296
#define OFF_WS    63488    // staged weight 64 x (LDW=65) = 4160 floats
#define OFF_CHX   67648    // per-channel |x2| partial sums (64)
#define OFF_CHY   67712    // per-channel |y2| partial sums (64)
#define LDS_FLOATS 67776   // 271,104 bytes

__device__ __forceinline__ v8f wmma4(v2f a, v2f b, v8f c) {
  // V_WMMA_F32_16X16X4_F32: D = A(16x4 f32) x B(4x16 f32) + C(16x16 f32)
  return __builtin_amdgcn_wmma_f32_16x16x4_f32(false, a, false, b, (short)0, c,
                                               false, false);
}

__launch_bounds__(256)
__global__ void scam_row_kernel(
    const float* __restrict__ x, const float* __restrict__ y,
    const float* __restrict__ ln_l_w, const float* __restrict__ ln_l_b,
    const float* __restrict__ ln_r_w, const float* __restrict__ ln_r_b,
    const float* __restrict__ Wl1, const float* __restrict__ bl1,
    const float* __restrict__ Wr1, const float* __restrict__ br1,
    const float* __restrict__ Wl2, const float* __restrict__ bl2,
    const float* __restrict__ Wr2, const float* __restrict__ br2,
    const float* __restrict__ beta, const float* __restrict__ gamma,
    float* __restrict__ x2, float* __restrict__ y2,
    float* __restrict__ r1, float* __restrict__ r2)
{
  extern __shared__ float s[];
  const int bh = blockIdx.x;
  const int b  = bh / H192;
  const int h  = bh % H192;
  const int tid  = threadIdx.x;
  const int wave = tid >> 5;
  const int lane = tid & 31;
  const int m    = lane & 15;          // row/col within fragment
  const int kb2  = (lane >> 4) * 2;    // K pair selector for A/B frags
  const int vrow = (lane >> 4) << 3;   // +8 row offset for C frags, lanes 16-31
  const size_t baseX = (size_t)(b * C64) * PLANE + (size_t)h * W192;

  if (tid < 128) s[OFF_CHX + tid] = 0.0f;   // zero CHX+CHY

  // ---------------- Phase 1: load rows + channel LayerNorm (in place) -------
  if (tid < W192) {
    const int w = tid;
    float sx = 0.f, sxx = 0.f, sy = 0.f, syy = 0.f;
    for (int c = 0; c < C64; ++c) {
      float xv = x[baseX + (size_t)c * PLANE + w];
      float yv = y[baseX + (size_t)c * PLANE + w];
      s[OFF_LNX + c * LDQ + w] = xv;
      s[OFF_LNY + c * LDQ + w] = yv;
      sx += xv; sxx += xv * xv;
      sy += yv; syy += yv * yv;
    }
    const float mux = sx * (1.f / 64.f), muy = sy * (1.f / 64.f);
    const float vx = sxx * (1.f / 64.f) - mux * mux;
    const float vy = syy * (1.f / 64.f) - muy * muy;
    const float rx = rsqrtf(vx + 1e-5f), ry = rsqrtf(vy + 1e-5f);
    for (int c = 0; c < C64; ++c) {
      s[OFF_LNX + c * LDQ + w] =
          (s[OFF_LNX + c * LDQ + w] - mux) * rx * ln_l_w[c] + ln_l_b[c];
      s[OFF_LNY + c * LDQ + w] =
          (s[OFF_LNY + c * LDQ + w] - muy) * ry * ln_r_w[c] + ln_r_b[c];
    }
  }
  __syncthreads();

  // helpers -----------------------------------------------------------------
  auto stageW = [&](const float* Wm) {
    for (int i = tid; i < 64 * 64; i += 256)
      s[OFF_WS + (i >> 6) * LDW + (i & 63)] = Wm[i];
  };
  // out[o][w] = sum_c WS[o][c] * LDS_src[c][w] + bias[o]
  auto gemm_lds = [&](int srcOff, int dstOff, const float* bias) {
    for (int t = wave; t < 48; t += 8) {
      const int row0 = (t & 3) * 16, col0 = (t >> 2) * 16;
      v8f acc = {};
      for (int k0 = 0; k0 < 64; k0 += 4) {
        const int k = k0 + kb2;
        v2f a, bf;
        a.x  = s[OFF_WS + (row0 + m) * LDW + k];
        a.y  = s[OFF_WS + (row0 + m) * LDW + k + 1];
        bf.x = s[srcOff + k * LDQ + col0 + m];
        bf.y = s[srcOff + (k + 1) * LDQ + col0 + m];
        acc = wmma4(a, bf, acc);
      }
      const int col = col0 + m;
#pragma unroll
      for (int v = 0; v < 8; ++v) {
        const int row = row0 + v + vrow;
        s[dstOff + row * LDQ + col] = acc[v] + bias[row];
      }
    }
  };
  // out[o][w] = sum_c WS[o][c] * src_global[c][w] + bias[o]   (src = x or y row)
  auto gemm_glb = [&](const float* src, int dstOff, const float* bias) {
    for (int t = wave; t < 48; t += 8) {
      const int row0 = (t & 3) * 16, col0 = (t >> 2) * 16;
      v8f acc = {};
      for (int k0 = 0; k0 < 64; k0 += 4) {
        const int k = k0 + kb2;
        v2f a, bf;
        a.x  = s[OFF_WS + (row0 + m) * LDW + k];
        a.y  = s[OFF_WS + (row0 + m) * LDW + k + 1];
        bf.x = src[baseX + (size_t)k * PLANE + col0 + m];
        bf.y = src[baseX + (size_t)(k + 1) * PLANE + col0 + m];
        acc = wmma4(a, bf, acc);
      }
      const int col = col0 + m;
#pragma unroll
      for (int v = 0; v < 8; ++v) {
        const int row = row0 + v + vrow;
        s[dstOff + row * LDQ + col] = acc[v] + bias[row];
      }
    }
  };

  // ---------------- Phase 2: Ql, Qr ----------------------------------------
  stageW(Wl1); __syncthreads();
  gemm_lds(OFF_LNX, OFF_QL, bl1); __syncthreads();
  stageW(Wr1); __syncthreads();
  gemm_lds(OFF_LNY, OFF_QR, br1); __syncthreads();

  // ---------------- Phase 3: attn[i][j] = 0.125 * sum_c Ql[c][i] Qr[c][j] --
  for (int t = wave; t < 144; t += 8) {
    const int i0 = (t % 12) * 16, j0 = (t / 12) * 16;
    v8f acc = {};
    for (int k0 = 0; k0 < 64; k0 += 4) {
      const int k = k0 + kb2;
      v2f a, bf;
      a.x  = s[OFF_QL + k * LDQ + i0 + m];        // A = Ql^T (transposed load)
      a.y  = s[OFF_QL + (k + 1) * LDQ + i0 + m];
      bf.x = s[OFF_QR + k * LDQ + j0 + m];
      bf.y = s[OFF_QR + (k + 1) * LDQ + j0 + m];
      acc = wmma4(a, bf, acc);
    }
#pragma unroll
    for (int v = 0; v < 8; ++v) {
      const int i = i0 + v + vrow;
      s[OFF_ATT + i * LDA + j0 + m] = acc[v] * 0.125f;   // scale = C^-0.5
    }
  }
  __syncthreads();

  // ---------------- Phase 4: softmax stats (row over j, col over i) --------
  if (tid < W192) {
    const int i = tid;
    float mx = -1e30f;
    for (int j = 0; j < W192; ++j) mx = fmaxf(mx, s[OFF_ATT + i * LDA + j]);
    float sm = 0.f;
    for (int j = 0; j < W192; ++j) sm += expf(s[OFF_ATT + i * LDA + j] - mx);
    s[OFF_RMAX + i] = mx; s[OFF_RSUM + i] = sm;
    const int j = tid;
    float cm = -1e30f;
    for (int ii = 0; ii < W192; ++ii) cm = fmaxf(cm, s[OFF_ATT + ii * LDA + j]);
    float cs = 0.f;
    for (int ii = 0; ii < W192; ++ii) cs += expf(s[OFF_ATT + ii * LDA + j] - cm);
    s[OFF_CMAX + j] = cm; s[OFF_CSUM + j] = cs;
  }
  __syncthreads();

  // ---------------- Phase 5: Vl, Vr (overwrite Ql/Qr regions) --------------
  stageW(Wl2); __syncthreads();
  gemm_glb(x, OFF_QL, bl2); __syncthreads();     // OFF_QL now holds Vl
  stageW(Wr2); __syncthreads();
  gemm_glb(y, OFF_QR, br2); __syncthreads();     // OFF_QR now holds Vr

  // ---------------- Phase 6: F_r2l -> x2 = 2x + beta * softmax_row @ Vr^T --
  // out tile [c][i]; A[c][j] = Vr[c][j]; B[j][i] = exp(att[i][j]-rowmax[i])
  for (int t = wave; t < 48; t += 8) {
    const int c0 = (t & 3) * 16, i0 = (t >> 2) * 16;
    const int i = i0 + m;
    const float rmx = s[OFF_RMAX + i];
    v8f acc = {};
    for (int k0 = 0; k0 < W192; k0 += 4) {
      const int k = k0 + kb2;                      // k = j
      v2f a, bf;
      a.x  = s[OFF_QR + (c0 + m) * LDQ + k];       // Vr
      a.y  = s[OFF_QR + (c0 + m) * LDQ + k + 1];
      bf.x = expf(s[OFF_ATT + i * LDA + k] - rmx); // exp applied in-frag
      bf.y = expf(s[OFF_ATT + i * LDA + k + 1] - rmx);
      acc = wmma4(a, bf, acc);
    }
    const float rs = 1.0f / s[OFF_RSUM + i];
#pragma unroll
    for (int v = 0; v < 8; ++v) {
      const int c = c0 + v + vrow;
      const size_t gi = baseX + (size_t)c * PLANE + i;
      const float o = 2.0f * x[gi] + beta[c] * acc[v] * rs;
      x2[gi] = o;
      atomicAdd(&s[OFF_CHX + c], fabsf(o));
    }
  }

  // ---------------- Phase 7: F_l2r -> y2 = 2y + gamma * softmax_col^T @ Vl -
  // out tile [c][j]; A[c][i] = Vl[c][i]; B[i][j] = exp(att[i][j]-colmax[j])
  for (int t = wave; t < 48; t += 8) {
    const int c0 = (t & 3) * 16, j0 = (t >> 2) * 16;
    const int j = j0 + m;
    const float cmx = s[OFF_CMAX + j];
    v8f acc = {};
    for (int k0 = 0; k0 < W192; k0 += 4) {
      const int k = k0 + kb2;                      // k = i
      v2f a, bf;
      a.x  = s[OFF_QL + (c0 + m) * LDQ + k];       // Vl
      a.y  = s[OFF_QL + (c0 + m) * LDQ + k + 1];
      bf.x = expf(s[OFF_ATT + k * LDA + j] - cmx);
      bf.y = expf(s[OFF_ATT + (k + 1) * LDA + j] - cmx);
      acc = wmma4(a, bf, acc);
    }
    const float cs = 1.0f / s[OFF_CSUM + j];
#pragma unroll
    for (int v = 0; v < 8; ++v) {
      const int c = c0 + v + vrow;
      const size_t gi = baseX + (size_t)c * PLANE + j;
      const float o = 2.0f * y[gi] + gamma[c] * acc[v] * cs;
      y2[gi] = o;
      atomicAdd(&s[OFF_CHY + c], fabsf(o));
    }
  }
  __syncthreads();

  // ---------------- flush per-channel L1 partial sums ----------------------
  if (tid < 64) {
    atomicAdd(&r1[b * 64 + tid], s[OFF_CHX + tid]);
    atomicAdd(&r2[b * 64 + tid], s[OFF_CHY + tid]);
  }
}

__global__ void zero_kernel(float* __restrict__ p) {
  p[blockIdx.x * blockDim.x + threadIdx.x] = 0.0f;
}

__global__ void fuse_kernel(const float* __restrict__ x2, const float* __restrict__ y2,
                            const float* __restrict__ r1, const float* __restrict__ r2,
                            float* __restrict__ out) {
  __shared__ float w1s[64], w2s[64];
  const int b   = blockIdx.x / 144;
  const int pix = (blockIdx.x % 144) * 256 + threadIdx.x;   // 0..36863
  if (threadIdx.x < 64) {
    const int c = threadIdx.x;
    const float e1 = expf(r1[b * 64 + c] * (1.0f / 36864.0f));
    const float e2 = expf(r2[b * 64 + c] * (1.0f / 36864.0f));
    const float rd = e1 + e2 + 1e-5f;
    w1s[c] = e1 / rd; w2s[c] = e2 / rd;
  }
  __syncthreads();
  const size_t base = (size_t)b * 64 * PLANE + pix;
  float a1 = 0.f, a2 = 0.f;
  for (int c = 0; c < 64; ++c) {
    a1 += fabsf(x2[base + (size_t)c * PLANE]);
    a2 += fabsf(y2[base + (size_t)c * PLANE]);
  }
  const float e1 = expf(a1 * (1.0f / 64.0f));
  const float e2 = expf(a2 * (1.0f / 64.0f));
  const float cd = e1 + e2 + 1e-5f;
  const float cw1 = e1 / cd, cw2 = e2 / cd;
  for (int c = 0; c < 64; ++c) {
    const size_t gi = base + (size_t)c * PLANE;
    out[gi] = (w1s[c] + cw1) * x2[gi] + (w2s[c] + cw2) * y2[gi];
  }
}

extern "C" void kernel_launch(void* const* d_in, const int* in_sizes, int n_in,
                              void* d_out, int out_size, void* d_ws, size_t ws_size,
                              hipStream_t stream) {
  const float* x      = (const float*)d_in[0];
  const float* y      = (const float*)d_in[1];
  const float* ln_l_w = (const float*)d_in[2];
  const float* ln_l_b = (const float*)d_in[3];
  const float* ln_r_w = (const float*)d_in[4];
  const float* ln_r_b = (const float*)d_in[5];
  const float* Wl1    = (const float*)d_in[6];
  const float* bl1    = (const float*)d_in[7];
  const float* Wr1    = (const float*)d_in[8];
  const float* br1    = (const float*)d_in[9];
  const float* Wl2    = (const float*)d_in[10];
  const float* bl2    = (const float*)d_in[11];
  const float* Wr2    = (const float*)d_in[12];
  const float* br2    = (const float*)d_in[13];
  const float* beta   = (const float*)d_in[14];
  const float* gamma  = (const float*)d_in[15];

  float* ws = (float*)d_ws;
  const size_t TENSOR = (size_t)4 * 64 * PLANE;   // 9,437,184 floats
  float* x2 = ws;
  float* y2 = ws + TENSOR;
  float* r1 = ws + 2 * TENSOR;
  float* r2 = r1 + 256;

  zero_kernel<<<1, 512, 0, stream>>>(r1);          // zeros r1 (256) + r2 (256)

  const size_t shmem = (size_t)LDS_FLOATS * sizeof(float);   // 271,104 B
  (void)hipFuncSetAttribute((const void*)scam_row_kernel,
                            hipFuncAttributeMaxDynamicSharedMemorySize, (int)shmem);
  scam_row_kernel<<<dim3(4 * 192), dim3(256), shmem, stream>>>(
      x, y, ln_l_w, ln_l_b, ln_r_w, ln_r_b, Wl1, bl1, Wr1, br1,
      Wl2, bl2, Wr2, br2, beta, gamma, x2, y2, r1, r2);

  fuse_kernel<<<dim3(4 * 144), dim3(256), 0, stream>>>(x2, y2, r1, r2, (float*)d_out);
}